// RandISH_31542239821859
// MI455X (gfx1250) — compile-verified
//
#include <hip/hip_runtime.h>

typedef __attribute__((ext_vector_type(2))) float v2f;
typedef __attribute__((ext_vector_type(4))) float v4f;
typedef __attribute__((ext_vector_type(8))) float v8f;

#define RAND_N 32
#define PLEG   10
#define CROW   20   // per-basis constant row: 10 coef*y0n, 9 one-hot*cscale, 1 slope

// Math: rotated r=(x,y,z) is unit, so cos(theta)=z, sin(theta)=-sqrt(x^2+y^2),
//   sin(theta)^d * cos(d*phi) = Re[ (-(x+iy))^d ]   (no sqrt / trig needed)
//   y0 = y0_norm * sum_p coef[p] z^p                (Horner, premultiplied)
//   al = exp2( -0.5*log2(e)*d(d+1) * tb ),  tb = 1/(1/(rough+eps)+eps)
//
// WMMA mapping (V_WMMA_F32_16X16X4_F32): A = 128x4 block of all 32 rotation
// matrices (4 rows per basis: x,y,z,pad), B = 4x16 block of 16 batch vectors.
// Outer loop f selects the 16-row slice; C/D layout gives each lane complete
// (x,y,z) triples for 2 bases of its batch column.
__global__ __launch_bounds__(256) void ish_wmma_kernel(
    const float* __restrict__ vec,       // B x 3
    const float* __restrict__ rough,     // B
    const float* __restrict__ matrices,  // 32 x 3 x 3
    const float* __restrict__ coeffs,    // 32 x 10
    const float* __restrict__ cscale,    // 32
    const float* __restrict__ y0n,       // 32
    const int*   __restrict__ degs,      // 32
    float* __restrict__ out,             // B x 64
    int nbatch)
{
    __shared__ float sMat[RAND_N * 9];
    __shared__ float sCoef[RAND_N * CROW];   // 80B rows, 16B aligned

    const int t = threadIdx.x;

    for (int i = t; i < RAND_N * 9; i += blockDim.x) sMat[i] = matrices[i];
    for (int i = t; i < RAND_N * CROW; i += blockDim.x) {
        int n = i / CROW, p = i - CROW * n;
        int dg = degs[n];
        float val;
        if (p < PLEG) {
            val = y0n[n] * coeffs[n * PLEG + p];        // premultiplied Legendre
        } else if (p < 19) {
            val = ((p - 9) == dg) ? cscale[n] : 0.0f;   // one-hot power weight
        } else {
            float d = (float)dg;
            val = -0.72134752f * d * (d + 1.0f);        // -0.5*log2(e)*d(d+1)
        }
        sCoef[i] = val;
    }
    __syncthreads();

    const int lane = t & 31;
    const int half = lane >> 4;
    const int col  = lane & 15;

    // Wave-uniform (SGPR) loop bounds: scalar loop control, EXEC stays all-1s
    // through the WMMA as the ISA requires.
    const int waveGlobal = __builtin_amdgcn_readfirstlane(
        blockIdx.x * (blockDim.x >> 5) + (t >> 5));
    const int nWaves = gridDim.x * (blockDim.x >> 5);
    const int ntiles = nbatch >> 4;

#pragma unroll 1
    for (int f = 0; f < 8; ++f) {
        // A fragment for rows m = 16f..16f+15:  A[m][k] = matrices[m>>2][k][m&3]
        // lanes 0-15 hold K0/K1, lanes 16-31 hold K2/K3.
        const int m  = 16 * f + col;
        const int nm = m >> 2, j = m & 3;
        float a0 = 0.0f, a1 = 0.0f;
        if (j < 3) {
            if (half == 0) { a0 = sMat[nm * 9 + j]; a1 = sMat[nm * 9 + 3 + j]; }
            else           { a0 = sMat[nm * 9 + 6 + j]; }
        }
        v2f afrag; afrag.x = a0; afrag.y = a1;

        // per-lane basis indices handled in this f slice (constants hoisted
        // by LICM into ~40 VGPRs: 10x ds_load_b128, once per f per wave)
        const int na = 4 * f + 2 * half;      // q = 0
        const float* __restrict__ cfa = &sCoef[na * CROW];
        const float* __restrict__ cfb = &sCoef[(na + 1) * CROW];

#pragma unroll 1
        for (int tile = waveGlobal; tile < ntiles; tile += nWaves) {
            const int b = tile * 16 + col;

            __builtin_prefetch(vec + (size_t)(tile + nWaves) * 48, 0, 1);

            const float* vp = vec + (size_t)3 * b;
            const float vx = vp[0], vy = vp[1], vz = vp[2];
            const float rg = rough[b];
            const float kap = __builtin_amdgcn_rcpf(rg + 1e-8f);
            const float tb  = __builtin_amdgcn_rcpf(kap + 1e-8f);

            v2f bfrag;
            if (half == 0) { bfrag.x = vx; bfrag.y = vy; }
            else           { bfrag.x = vz; bfrag.y = 0.0f; }

            v8f c = {};
            v8f d = __builtin_amdgcn_wmma_f32_16x16x4_f32(
                false, afrag, false, bfrag, (short)0, c, false, false);

            v4f res;
#pragma unroll
            for (int q = 0; q < 2; ++q) {
                const float* __restrict__ cf = q ? cfb : cfa;
                const float x = d[4 * q + 0];
                const float y = d[4 * q + 1];
                const float z = d[4 * q + 2];

                // y0 (premultiplied Horner)
                float v = cf[9];
#pragma unroll
                for (int p = 8; p >= 0; --p) v = __builtin_fmaf(v, z, cf[p]);

                // yl1 = sum_m onehot[m] * Re[(-(x+iy))^m], mask-free
                const float wr = -x, wi = -y;
                float pr = wr, pi = wi;
                float acc = cf[10] * pr;
#pragma unroll
                for (int mm = 2; mm <= 9; ++mm) {
                    float t1 = pi * wi;
                    float t2 = pi * wr;
                    float nr = __builtin_fmaf(pr, wr, -t1);
                    float ni = __builtin_fmaf(pr, wi,  t2);
                    pr = nr; pi = ni;
                    acc = __builtin_fmaf(cf[9 + mm], pr, acc);
                }

                const float al = __builtin_amdgcn_exp2f(cf[19] * tb);
                res[2 * q + 0] = al * v;
                res[2 * q + 1] = al * acc;
            }

            // out[b][8f+4h .. +3] : one 16B store per (f, tile)
            *(v4f*)(out + (size_t)b * 64 + 8 * f + 4 * half) = res;
        }
    }
}

extern "C" void kernel_launch(void* const* d_in, const int* in_sizes, int n_in,
                              void* d_out, int out_size, void* d_ws, size_t ws_size,
                              hipStream_t stream) {
    const float* vec      = (const float*)d_in[0];
    const float* rough    = (const float*)d_in[1];
    const float* matrices = (const float*)d_in[2];
    const float* coeffs   = (const float*)d_in[3];
    const float* cscale   = (const float*)d_in[4];
    const float* y0n      = (const float*)d_in[5];
    const int*   degrees  = (const int*)d_in[6];
    float* out = (float*)d_out;

    const int nbatch = in_sizes[1];  // roughness has B elements

    dim3 grid(512), block(256);      // 4096 waves, 4 tiles each for B=262144
    hipLaunchKernelGGL(ish_wmma_kernel, grid, block, 0, stream,
                       vec, rough, matrices, coeffs, cscale, y0n, degrees, out, nbatch);
}